// S4_with_shared_A_5823975653751
// MI455X (gfx1250) — compile-verified
//
#include <hip/hip_runtime.h>
#include <math.h>

#define D_CH 2048
#define NS   64
#define LSEQ 8192
#define TCH  64
#define JC   (LSEQ / TCH)   // 128 chunks

typedef __attribute__((ext_vector_type(2))) float v2f;
typedef __attribute__((ext_vector_type(8))) float v8f;

// workspace layout (float offsets)
#define WS_DA    0          // dA               (4096)
#define WS_DAT   4096       // dA^64            (4096)
#define WS_VI    8192       // inv(I-d/2 A)     (4096)
#define WS_DELTA 12288      // delta scalar     (16)
#define WS_SEND  12304      // chunk end states (64*128)
#define WS_CST   20496      // chunk in-states  (64*128)
#define WS_DB    28688      // dB               (64*2048)
#define WS_U     159760     // U -> H in place  (64*8192)
// total = 684048 floats = 2.74 MB

// ---------------------------------------------------------------------------
// Kernel 1: discretize. Gauss-Jordan inverse of (I - d/2 A) in LDS, then
// dA = inv @ (I + d/2 A), and dA^64 via 6 squarings (for the carry phase).
// ---------------------------------------------------------------------------
__global__ __launch_bounds__(256) void setup_kernel(
    const float* __restrict__ A, const float* __restrict__ log_delta,
    float* __restrict__ ws) {
  __shared__ float sM[64][64];
  __shared__ float sV[64][64];
  __shared__ float sT[64][64];
  __shared__ float sF[64];
  const int tid = threadIdx.x;
  const float delta = __expf(log_delta[0]);
  const float hd = 0.5f * delta;

  for (int idx = tid; idx < 4096; idx += 256) {
    int r = idx >> 6, c = idx & 63;
    float a = A[idx];
    sM[r][c] = (r == c ? 1.0f : 0.0f) - hd * a;
    sV[r][c] = (r == c ? 1.0f : 0.0f);
  }
  __syncthreads();

  // Gauss-Jordan; diagonally dominant -> no pivoting needed
  for (int k = 0; k < 64; ++k) {
    float invp = 1.0f / sM[k][k];
    __syncthreads();
    if (tid < 64)       sM[k][tid] *= invp;
    else if (tid < 128) sV[k][tid - 64] *= invp;
    __syncthreads();
    if (tid < 64) sF[tid] = sM[tid][k];
    __syncthreads();
    for (int idx = tid; idx < 4096; idx += 256) {
      int r = idx >> 6, c = idx & 63;
      if (r != k) {
        float f = sF[r];
        sM[r][c] -= f * sM[k][c];
        sV[r][c] -= f * sV[k][c];
      }
    }
    __syncthreads();
  }

  // dA = inv @ (I + hd*A) = inv + hd*(inv @ A)
  for (int idx = tid; idx < 4096; idx += 256) {
    int r = idx >> 6, c = idx & 63;
    float s = 0.0f;
    for (int k = 0; k < 64; ++k) s += sV[r][k] * A[k * 64 + c];
    s = sV[r][c] + hd * s;
    sT[r][c] = s;
    ws[WS_DA + idx] = s;
    ws[WS_VI + idx] = sV[r][c];
  }
  if (tid == 0) ws[WS_DELTA] = delta;
  __syncthreads();

  // dA^64 by 6 squarings
  for (int it = 0; it < 6; ++it) {
    for (int idx = tid; idx < 4096; idx += 256) {
      int r = idx >> 6, c = idx & 63;
      float s = 0.0f;
      for (int k = 0; k < 64; ++k) s += sT[r][k] * sT[k][c];
      sM[r][c] = s;
    }
    __syncthreads();
    for (int idx = tid; idx < 4096; idx += 256)
      sT[idx >> 6][idx & 63] = sM[idx >> 6][idx & 63];
    __syncthreads();
  }
  for (int idx = tid; idx < 4096; idx += 256)
    ws[WS_DAT + idx] = sT[idx >> 6][idx & 63];
}

// ---------------------------------------------------------------------------
// Kernel 2: dB = delta * inv @ B   (64 x 2048, 16.8 MFLOP)
// ---------------------------------------------------------------------------
__global__ __launch_bounds__(256) void dB_kernel(
    const float* __restrict__ B, float* __restrict__ ws) {
  const float delta = ws[WS_DELTA];
  int gid = blockIdx.x * 256 + threadIdx.x;
  for (int o = gid; o < NS * D_CH; o += 64 * 256) {
    int n = o >> 11;
    int d = o & (D_CH - 1);
    const float* vi = ws + WS_VI + n * 64;
    float s = 0.0f;
    for (int k = 0; k < 64; ++k) s += vi[k] * B[k * D_CH + d];
    ws[WS_DB + o] = delta * s;
  }
}

// ---------------------------------------------------------------------------
// Kernel 3: U = dB @ X  (64x2048)@(2048x8192), f32 WMMA 16x16x4.
// Block = 8 waves; each wave owns 16 columns and all 4 M-tiles.
// ---------------------------------------------------------------------------
__global__ __launch_bounds__(256) void gemm1_kernel(
    const float* __restrict__ X, float* __restrict__ ws) {
  const float* __restrict__ dB = ws + WS_DB;
  float* __restrict__ U = ws + WS_U;
  const int lane = threadIdx.x & 31;
  const int wave = threadIdx.x >> 5;
  const int m    = lane & 15;
  const int col  = blockIdx.x * 128 + wave * 16 + m;
  const int koff = (lane >> 4) << 1;  // 0 or 2

  v8f acc0 = 0.0f, acc1 = 0.0f, acc2 = 0.0f, acc3 = 0.0f;
  for (int k = 0; k < D_CH; k += 4) {
    v2f b;
    b.x = X[(k + koff) * LSEQ + col];
    b.y = X[(k + koff + 1) * LSEQ + col];
    const float* dbp = dB + k + koff;
    v2f a0 = *(const v2f*)(dbp + (m +  0) * D_CH);
    v2f a1 = *(const v2f*)(dbp + (m + 16) * D_CH);
    v2f a2 = *(const v2f*)(dbp + (m + 32) * D_CH);
    v2f a3 = *(const v2f*)(dbp + (m + 48) * D_CH);
    acc0 = __builtin_amdgcn_wmma_f32_16x16x4_f32(false, a0, false, b, (short)0, acc0, false, false);
    acc1 = __builtin_amdgcn_wmma_f32_16x16x4_f32(false, a1, false, b, (short)0, acc1, false, false);
    acc2 = __builtin_amdgcn_wmma_f32_16x16x4_f32(false, a2, false, b, (short)0, acc2, false, false);
    acc3 = __builtin_amdgcn_wmma_f32_16x16x4_f32(false, a3, false, b, (short)0, acc3, false, false);
  }
  const int ro = (lane >> 4) << 3;  // 0 or 8
  #pragma unroll
  for (int v = 0; v < 8; ++v) {
    U[( 0 + v + ro) * LSEQ + col] = acc0[v];
    U[(16 + v + ro) * LSEQ + col] = acc1[v];
    U[(32 + v + ro) * LSEQ + col] = acc2[v];
    U[(48 + v + ro) * LSEQ + col] = acc3[v];
  }
}

// ---------------------------------------------------------------------------
// Kernel 4: chunk-local scans (in place over U), record local end states.
// sAT[k][n] = dA[n][k]; pad 65 -> conflict-free LDS.
// ---------------------------------------------------------------------------
__global__ __launch_bounds__(64) void scan_kernel(float* __restrict__ ws) {
  __shared__ float sAT[64][65];
  __shared__ float h[64];
  const float* __restrict__ dA = ws + WS_DA;
  float* __restrict__ U = ws + WS_U;
  const int n = threadIdx.x;
  const int j = blockIdx.x;
  for (int it = 0; it < 64; ++it) sAT[n][it] = dA[it * 64 + n];
  h[n] = 0.0f;
  __syncthreads();
  float* up = U + n * LSEQ + j * TCH;
  for (int i = 0; i < TCH; ++i) {
    float s = up[i];
    #pragma unroll 8
    for (int k = 0; k < 64; ++k) s += sAT[k][n] * h[k];
    __syncthreads();
    h[n] = s;
    up[i] = s;
    __syncthreads();
  }
  ws[WS_SEND + j * 64 + n] = h[n];
}

// ---------------------------------------------------------------------------
// Kernel 5: sequential carry over 128 chunks: c_{j+1} = dA^64 c_j + send_j
// ---------------------------------------------------------------------------
__global__ __launch_bounds__(64) void carry_kernel(float* __restrict__ ws) {
  __shared__ float sAT[64][65];
  __shared__ float h[64];
  const float* __restrict__ dAT = ws + WS_DAT;
  const int n = threadIdx.x;
  for (int it = 0; it < 64; ++it) sAT[n][it] = dAT[it * 64 + n];
  h[n] = 0.0f;
  __syncthreads();
  for (int j = 0; j < JC; ++j) {
    ws[WS_CST + j * 64 + n] = h[n];
    float s = ws[WS_SEND + j * 64 + n];
    #pragma unroll 8
    for (int k = 0; k < 64; ++k) s += sAT[k][n] * h[k];
    __syncthreads();
    h[n] = s;
    __syncthreads();
  }
}

// ---------------------------------------------------------------------------
// Kernel 6: fix-up: H[:, jT+i] += dA^{i+1} c_j   (in place over U)
// ---------------------------------------------------------------------------
__global__ __launch_bounds__(64) void fixup_kernel(float* __restrict__ ws) {
  __shared__ float sAT[64][65];
  __shared__ float h[64];
  const float* __restrict__ dA = ws + WS_DA;
  float* __restrict__ U = ws + WS_U;
  const int n = threadIdx.x;
  const int j = blockIdx.x;
  for (int it = 0; it < 64; ++it) sAT[n][it] = dA[it * 64 + n];
  h[n] = ws[WS_CST + j * 64 + n];
  __syncthreads();
  float* up = U + n * LSEQ + j * TCH;
  for (int i = 0; i < TCH; ++i) {
    float s = 0.0f;
    #pragma unroll 8
    for (int k = 0; k < 64; ++k) s += sAT[k][n] * h[k];
    __syncthreads();
    h[n] = s;
    up[i] += s;
    __syncthreads();
  }
}

// ---------------------------------------------------------------------------
// Kernel 7: Y = C @ H + Dp .* X  (2048x64)@(64x8192), f32 WMMA, fused skip.
// Block = 8 waves; wave owns 16 rows x 64 cols (4 N-tiles).
// ---------------------------------------------------------------------------
__global__ __launch_bounds__(256) void gemm2_kernel(
    const float* __restrict__ X, const float* __restrict__ Cm,
    const float* __restrict__ Dp, const float* __restrict__ ws,
    float* __restrict__ Y) {
  const float* __restrict__ H = ws + WS_U;
  const int lane  = threadIdx.x & 31;
  const int wave  = threadIdx.x >> 5;
  const int cl    = lane & 15;
  const int rbase = blockIdx.y * 128 + wave * 16;
  const int cbase = blockIdx.x * 64;
  const int koff  = (lane >> 4) << 1;

  v8f acc0 = 0.0f, acc1 = 0.0f, acc2 = 0.0f, acc3 = 0.0f;
  #pragma unroll
  for (int k = 0; k < NS; k += 4) {
    v2f a = *(const v2f*)(Cm + (rbase + cl) * NS + k + koff);
    const float* h0 = H + (k + koff) * LSEQ + cbase;
    const float* h1 = H + (k + koff + 1) * LSEQ + cbase;
    v2f b0, b1, b2, b3;
    b0.x = h0[cl];      b0.y = h1[cl];
    b1.x = h0[16 + cl]; b1.y = h1[16 + cl];
    b2.x = h0[32 + cl]; b2.y = h1[32 + cl];
    b3.x = h0[48 + cl]; b3.y = h1[48 + cl];
    acc0 = __builtin_amdgcn_wmma_f32_16x16x4_f32(false, a, false, b0, (short)0, acc0, false, false);
    acc1 = __builtin_amdgcn_wmma_f32_16x16x4_f32(false, a, false, b1, (short)0, acc1, false, false);
    acc2 = __builtin_amdgcn_wmma_f32_16x16x4_f32(false, a, false, b2, (short)0, acc2, false, false);
    acc3 = __builtin_amdgcn_wmma_f32_16x16x4_f32(false, a, false, b3, (short)0, acc3, false, false);
  }
  const int ro = (lane >> 4) << 3;
  #pragma unroll
  for (int v = 0; v < 8; ++v) {
    int d = rbase + v + ro;
    float dp = Dp[d];
    long off = (long)d * LSEQ + cbase + cl;
    Y[off     ] = acc0[v] + dp * X[off     ];
    Y[off + 16] = acc1[v] + dp * X[off + 16];
    Y[off + 32] = acc2[v] + dp * X[off + 32];
    Y[off + 48] = acc3[v] + dp * X[off + 48];
  }
}

extern "C" void kernel_launch(void* const* d_in, const int* in_sizes, int n_in,
                              void* d_out, int out_size, void* d_ws, size_t ws_size,
                              hipStream_t stream) {
  const float* X         = (const float*)d_in[0];
  const float* A         = (const float*)d_in[1];
  const float* B         = (const float*)d_in[2];
  const float* C         = (const float*)d_in[3];
  const float* Dp        = (const float*)d_in[4];
  const float* log_delta = (const float*)d_in[5];
  float* ws = (float*)d_ws;
  float* Y  = (float*)d_out;

  setup_kernel<<<1, 256, 0, stream>>>(A, log_delta, ws);
  dB_kernel<<<64, 256, 0, stream>>>(B, ws);
  gemm1_kernel<<<64, 256, 0, stream>>>(X, ws);       // 8192 cols / 128 per block
  scan_kernel<<<JC, 64, 0, stream>>>(ws);
  carry_kernel<<<1, 64, 0, stream>>>(ws);
  fixup_kernel<<<JC, 64, 0, stream>>>(ws);
  gemm2_kernel<<<dim3(128, 16), 256, 0, stream>>>(X, C, Dp, ws, Y);
}